// WARPLoss_39968965656935
// MI455X (gfx1250) — compile-verified
//
#include <hip/hip_runtime.h>
#include <cstdint>

// ---------------------------------------------------------------------------
// WARP loss for MI455X (gfx1250).
// Roofline: the only bulk term is one streaming scan of `target`
// (2048 x 100000 x 4B = 819 MB -> ~35 us @ 23.3 TB/s). Everything else is
// ~2048x51 scattered loads + O(1e5) FLOPs. No dense matrix math -> no WMMA;
// the CDNA5 path that matters is the Tensor Data Mover: a 4-deep ring of
// 16 KB tensor_load_to_lds DMAs (NT cache hint) synchronized with
// s_wait_tensorcnt, while 8 waves scan LDS for the one-hot index.
// ---------------------------------------------------------------------------

#define SCAN_CHUNK 4096          // ints per TDM tile (16 KB)
#define SCAN_BUFS  4             // LDS ring slots (64 KB total)
#define SCAN_PRE   3             // prefetch depth (chunks in flight ahead)
#define SCAN_BLOCK 256
#define TRIAL_BLOCK 64

typedef unsigned int v4u __attribute__((ext_vector_type(4)));
typedef int          v8i __attribute__((ext_vector_type(8)));
typedef int          v4i __attribute__((ext_vector_type(4)));

// cpol: TH in bits[2:0]; TH=1 -> non-temporal (819 MB stream > 192 MB L2).
#define TDM_CPOL_NT 1

// Build a 1-D TDM descriptor (D#) and issue TENSOR_LOAD_TO_LDS.
// group0: [1:0]=count=1, [63:32]=lds_addr, [120:64]=global_addr, [127:126]=type=2
// group1: [17:16]=data_size(2 -> 4B), [79:48]=tensor_dim0, [111:80]=tensor_dim1,
//         [127:112]=tile_dim0, [207:160]=tensor_dim0_stride
__device__ __forceinline__ void tdm_load_row_chunk(uint32_t lds_off,
                                                   const void* gptr,
                                                   uint32_t nelem) {
  uint64_t ga = (uint64_t)(uintptr_t)gptr;
  v4u g0;
  g0.x = 0x1u;                                    // count = 1 valid descriptor
  g0.y = lds_off;                                 // LDS byte address
  g0.z = (uint32_t)ga;                            // global_addr[31:0]
  g0.w = (uint32_t)(ga >> 32) | 0x80000000u;      // global_addr[56:32] | type=2<<30

  v8i g1;
  g1[0] = 0x00020000;                             // data_size = 2 (4 bytes)
  g1[1] = (int)((nelem & 0xFFFFu) << 16);         // tensor_dim0[15:0]
  g1[2] = (int)((nelem >> 16) & 0xFFFFu) |        // tensor_dim0[31:16]
          (1 << 16);                              // tensor_dim1 = 1
  g1[3] = (int)((nelem & 0xFFFFu) << 16);         // tile_dim0 = nelem
  g1[4] = 0;                                      // tile_dim1/2 unused
  g1[5] = (int)nelem;                             // tensor_dim0_stride[31:0]
  g1[6] = 0;
  g1[7] = 0;

  v4i z4 = {0, 0, 0, 0};
#if defined(__clang_major__) && (__clang_major__ >= 23)
  v8i z8 = {0, 0, 0, 0, 0, 0, 0, 0};
  __builtin_amdgcn_tensor_load_to_lds(g0, g1, z4, z4, z8, TDM_CPOL_NT);
#else
  __builtin_amdgcn_tensor_load_to_lds(g0, g1, z4, z4, TDM_CPOL_NT);
#endif
}

// s_wait_tensorcnt needs a constant immediate; dispatch the tail values.
__device__ __forceinline__ void wait_tensorcnt_le(int n) {
  switch (n) {
    case 0:  __builtin_amdgcn_s_wait_tensorcnt(0); break;
    case 1:  __builtin_amdgcn_s_wait_tensorcnt(1); break;
    case 2:  __builtin_amdgcn_s_wait_tensorcnt(2); break;
    default: __builtin_amdgcn_s_wait_tensorcnt(3); break;
  }
}

// Low-bias 32-bit integer hash (counter-based RNG for negative sampling).
__device__ __forceinline__ uint32_t hash32(uint32_t x) {
  x ^= x >> 16; x *= 0x7feb352dU;
  x ^= x >> 15; x *= 0x846ca68bU;
  x ^= x >> 16;
  return x;
}

// ---------------------------------------------------------------------------
// Kernel 1: find positive label index per row by scanning one-hot target.
// TDM streams 16 KB chunks into a 4-slot LDS ring (3 in flight); 8 waves
// scan int4 vectors from LDS.
// ---------------------------------------------------------------------------
__global__ void __launch_bounds__(SCAN_BLOCK)
find_pos_kernel(const int* __restrict__ target, int* __restrict__ pos, int Y) {
  __shared__ int buf[SCAN_BUFS][SCAN_CHUNK];
  const int row = blockIdx.x;
  const int* trow = target + (size_t)row * (size_t)Y;
  const int nc = (Y + SCAN_CHUNK - 1) / SCAN_CHUNK;

  // Scalar branch: only wave 0 issues TDM ops (TDM ignores EXEC).
  const bool is_wave0 = (__builtin_amdgcn_readfirstlane((int)threadIdx.x) < 32);

  if (is_wave0) {
    const int npre = (nc < SCAN_PRE) ? nc : SCAN_PRE;
    for (int i = 0; i < npre; ++i) {
      const int base = i * SCAN_CHUNK;
      const int n = (Y - base < SCAN_CHUNK) ? (Y - base) : SCAN_CHUNK;
      tdm_load_row_chunk((uint32_t)(uintptr_t)&buf[i][0], trow + base,
                         (uint32_t)n);
    }
  }

  for (int c = 0; c < nc; ++c) {
    const int base = c * SCAN_CHUNK;
    const int n = (Y - base < SCAN_CHUNK) ? (Y - base) : SCAN_CHUNK;

    if (is_wave0) {
      if (c + SCAN_PRE < nc) {
        const int base1 = (c + SCAN_PRE) * SCAN_CHUNK;
        const int n1 = (Y - base1 < SCAN_CHUNK) ? (Y - base1) : SCAN_CHUNK;
        // Slot (c+PRE)%BUFS == (c-1)%BUFS: freed at the previous iteration's
        // trailing barrier, so reuse is safe.
        tdm_load_row_chunk((uint32_t)(uintptr_t)&buf[(c + SCAN_PRE) % SCAN_BUFS][0],
                           trow + base1, (uint32_t)n1);
      }
      // Chunks issued so far = min(c+PRE+1, nc); require chunk c complete
      // (TDM ops from one wave complete in order).
      const int last_issued = (c + SCAN_PRE < nc) ? (c + SCAN_PRE) : (nc - 1);
      wait_tensorcnt_le(last_issued - c);
    }
    __syncthreads();                            // publish chunk c to all waves

    const int* s = &buf[c % SCAN_BUFS][0];
    const int nv4 = n >> 2;
    const int4* s4 = (const int4*)s;
    for (int i = threadIdx.x; i < nv4; i += SCAN_BLOCK) {
      int4 v = s4[i];
      if (v.x | v.y | v.z | v.w) {
        int idx = base + i * 4;
        if (v.x)      pos[row] = idx;
        else if (v.y) pos[row] = idx + 1;
        else if (v.z) pos[row] = idx + 2;
        else          pos[row] = idx + 3;
      }
    }
    for (int i = (nv4 << 2) + threadIdx.x; i < n; i += SCAN_BLOCK) {
      if (s[i]) pos[row] = base + i;
    }
    __syncthreads();                            // slot (c%BUFS) free for reuse
  }
}

// ---------------------------------------------------------------------------
// Kernel 2: WARP rejection-sampling trials. One 64-thread block per row;
// lane t handles trial t. First violating trial via LDS atomicMin.
// ---------------------------------------------------------------------------
__global__ void __launch_bounds__(TRIAL_BLOCK)
warp_trials_kernel(const float* __restrict__ inp, const int* __restrict__ pos,
                   const int* __restrict__ pT, float* __restrict__ loss, int Y) {
  __shared__ float negs[TRIAL_BLOCK];
  __shared__ int firstIdx;

  const int row = blockIdx.x;
  const int t = threadIdx.x;
  int T = *pT;
  if (T > TRIAL_BLOCK) T = TRIAL_BLOCK;

  if (t == 0) firstIdx = 0x7FFFFFFF;

  int j = pos[row];
  j = (j < 0) ? 0 : ((j >= Y) ? (Y - 1) : j);       // safety clamp
  const float ps = inp[(size_t)row * (size_t)Y + j];
  __syncthreads();

  if (t < T) {
    // uniform negative != j (counter-based RNG; collisions negligible, T<<Y)
    uint32_t h = hash32((uint32_t)(row * TRIAL_BLOCK + t) ^ 0x9E3779B9u);
    uint32_t idx = h % (uint32_t)(Y - 1);
    if ((int)idx >= j) idx++;
    const float ns = inp[(size_t)row * (size_t)Y + idx];
    negs[t] = ns;
    if (1.0f + ns - ps >= 0.0f) atomicMin(&firstIdx, t);
  }
  __syncthreads();

  if (t == 0) {
    float out = 0.0f;
    if (firstIdx < T) {
      const float nt = (float)(firstIdx + 1);
      const float L = __logf(floorf(((float)Y - 1.0f) / nt));
      out = L * (1.0f - ps + negs[firstIdx]);
    }
    loss[row] = out;
  }
}

// ---------------------------------------------------------------------------
// Kernel 3: deterministic tree reduction of per-row losses -> d_out[0].
// ---------------------------------------------------------------------------
__global__ void __launch_bounds__(256)
reduce_kernel(const float* __restrict__ loss, float* __restrict__ out, int B) {
  __shared__ float sm[256];
  float s = 0.0f;
  for (int i = threadIdx.x; i < B; i += 256) s += loss[i];
  sm[threadIdx.x] = s;
  __syncthreads();
  for (int w = 128; w > 0; w >>= 1) {
    if ((int)threadIdx.x < w) sm[threadIdx.x] += sm[threadIdx.x + w];
    __syncthreads();
  }
  if (threadIdx.x == 0) out[0] = sm[0];
}

// ---------------------------------------------------------------------------
extern "C" void kernel_launch(void* const* d_in, const int* in_sizes, int n_in,
                              void* d_out, int out_size, void* d_ws, size_t ws_size,
                              hipStream_t stream) {
  const float* inp    = (const float*)d_in[0];
  const int*   target = (const int*)d_in[1];
  const int*   pT     = (const int*)d_in[2];     // max_num_trials (device scalar)

  const int Y = 100000;
  const int B = in_sizes[0] / Y;

  int*   pos  = (int*)d_ws;
  float* loss = (float*)((char*)d_ws + (size_t)B * sizeof(int));

  find_pos_kernel<<<B, SCAN_BLOCK, 0, stream>>>(target, pos, Y);
  warp_trials_kernel<<<B, TRIAL_BLOCK, 0, stream>>>(inp, pos, pT, loss, Y);
  reduce_kernel<<<1, 256, 0, stream>>>(loss, (float*)d_out, B);
}